// GatedDeltaNetLayer_75892072120383
// MI455X (gfx1250) — compile-verified
//
#include <hip/hip_runtime.h>

typedef __attribute__((ext_vector_type(2))) float v2f;
typedef __attribute__((ext_vector_type(8))) float v8f;
typedef __attribute__((address_space(3))) const float* lds_cfp;

static constexpr int Bc = 2, Tc = 2048, Dc = 2048, Hc = 16, Kc = 128, Vc = 128;

// ---------------------------------------------------------------------------
// CDNA5 async global->LDS copy (16B per lane), tracked by ASYNCcnt.
// ---------------------------------------------------------------------------
__device__ __forceinline__ void async_cp_b128(const float* g, const float* l) {
  const unsigned lds_off = (unsigned)(size_t)(lds_cfp)l;          // addrspace(3) byte offset
  asm volatile("global_load_async_to_lds_b128 %0, %1, off"
               :: "v"(lds_off), "v"((unsigned long long)(size_t)g)
               : "memory");
}

// ---------------------------------------------------------------------------
// RMSNorm: one block per row of D=2048
// ---------------------------------------------------------------------------
__global__ __launch_bounds__(256) void rmsnorm_kernel(
    const float* __restrict__ x, const float* __restrict__ w,
    float* __restrict__ xn, int D) {
  __shared__ float red[8];
  const int r = blockIdx.x;
  const float* xr = x + (size_t)r * D;
  float ss = 0.f;
  for (int i = threadIdx.x; i < D; i += 256) {
    float t = xr[i];
    ss += t * t;
  }
#pragma unroll
  for (int off = 16; off; off >>= 1) ss += __shfl_xor(ss, off, 32);
  const int lane = threadIdx.x & 31, wv = threadIdx.x >> 5;
  if (lane == 0) red[wv] = ss;
  __syncthreads();
  if (wv == 0) {
    float t = (lane < 8) ? red[lane] : 0.f;
#pragma unroll
    for (int off = 4; off; off >>= 1) t += __shfl_xor(t, off, 32);
    if (lane == 0) red[0] = t;
  }
  __syncthreads();
  const float rinv = rsqrtf(red[0] / (float)D + 1e-6f);
  float* xo = xn + (size_t)r * D;
  for (int i = threadIdx.x; i < D; i += 256) xo[i] = xr[i] * rinv * w[i];
}

// ---------------------------------------------------------------------------
// f32 GEMM via V_WMMA_F32_16X16X4_F32 with double-buffered async-to-LDS
// staging. C[M,N] = A[M,K]*B[K,N] (+resid), row major.
// Block 256 threads = 8 waves (4M x 2N). Wave tile 32x64, block tile 128x128,
// BK = 32. LDS: 2 x (A 128x32 + B 32x128) f32 = 64 KB.
// ---------------------------------------------------------------------------
#define GEMM_BM 128
#define GEMM_BN 128
#define GEMM_BK 32

__global__ __launch_bounds__(256) void gemm_wmma_f32(
    const float* __restrict__ A, const float* __restrict__ Bm,
    const float* __restrict__ resid, float* __restrict__ C,
    int M, int N, int K) {
  __shared__ float sA[2][GEMM_BM * GEMM_BK];   // [row][k] row-major
  __shared__ float sB[2][GEMM_BK * GEMM_BN];   // [k][col] row-major

  const int tid = threadIdx.x;
  const int lane = tid & 31;
  const int wave = tid >> 5;
  const int wm = wave & 3, wn = wave >> 2;
  const int bm = blockIdx.y * GEMM_BM;
  const int bn = blockIdx.x * GEMM_BN;
  const int half = lane >> 4;   // 0: k0/k1 lanes, 1: k2/k3 lanes
  const int l16 = lane & 15;

  // Staging decomposition: A tile = 1024 x b128 chunks, B tile = 1024 chunks.
  const int ar = tid >> 3;            // A: 32 rows per pass (stride 32)
  const int ac = (tid & 7) * 4;       // A: 8 x float4 per 32-float row... (BK=32 -> 8 chunks/row)
  const int br = tid >> 6;            // B: 4 rows per pass (stride 4? no: 256/64) 
  const int bc = (tid & 63) * 2;      // placeholder (recomputed below)

  // Issue one chunk's async copies (8 x b128 per thread).
  auto stage = [&](int buf, int k0) {
#pragma unroll
    for (int it = 0; it < 4; ++it) {                 // A: 128 rows, 8 chunks/row
      const int r = ar + it * 32;
      async_cp_b128(A + (size_t)(bm + r) * K + k0 + ac,
                    &sA[buf][r * GEMM_BK + ac]);
    }
#pragma unroll
    for (int it = 0; it < 4; ++it) {                 // B: 32 rows, 32 chunks/row
      const int idx = tid + it * 256;                // 0..1023
      const int r = idx >> 5;                        // 0..31
      const int c = (idx & 31) * 4;                  // 0..124
      async_cp_b128(Bm + (size_t)(k0 + r) * N + bn + c,
                    &sB[buf][r * GEMM_BN + c]);
    }
  };

  (void)br; (void)bc;

  v8f acc[2][4] = {};
  const int nk = K / GEMM_BK;

  stage(0, 0);
  for (int kc = 0; kc < nk; ++kc) {
    const int cur = kc & 1;
    if (kc + 1 < nk) {
      stage(cur ^ 1, (kc + 1) * GEMM_BK);
      asm volatile("s_wait_asynccnt 0x8" ::: "memory");   // prev chunk landed
    } else {
      asm volatile("s_wait_asynccnt 0x0" ::: "memory");
    }
    __syncthreads();

    const float* __restrict__ pA = &sA[cur][0];
    const float* __restrict__ pB = &sB[cur][0];
#pragma unroll
    for (int kk = 0; kk < GEMM_BK; kk += 4) {
      v2f a[2];
#pragma unroll
      for (int i = 0; i < 2; ++i) {
        const int row = wm * 32 + i * 16 + l16;
        a[i] = *(const v2f*)(pA + row * GEMM_BK + kk + 2 * half);
      }
      v2f b[4];
#pragma unroll
      for (int j = 0; j < 4; ++j) {
        const int col = wn * 64 + j * 16 + l16;
        b[j].x = pB[(kk + 2 * half) * GEMM_BN + col];
        b[j].y = pB[(kk + 2 * half + 1) * GEMM_BN + col];
      }
#pragma unroll
      for (int i = 0; i < 2; ++i)
#pragma unroll
        for (int j = 0; j < 4; ++j)
          acc[i][j] = __builtin_amdgcn_wmma_f32_16x16x4_f32(
              false, a[i], false, b[j], (short)0, acc[i][j], false, false);
    }
    __syncthreads();   // all waves done reading before buffer is re-filled
  }

  // Epilogue: C/D layout — VGPR e holds row (e + 8*half), col = l16
  const int m0 = bm + wm * 32;
  const int n0 = bn + wn * 64;
#pragma unroll
  for (int i = 0; i < 2; ++i) {
#pragma unroll
    for (int j = 0; j < 4; ++j) {
#pragma unroll
      for (int e = 0; e < 8; ++e) {
        const int row = m0 + i * 16 + e + 8 * half;
        const int col = n0 + j * 16 + l16;
        float val = acc[i][j][e];
        if (resid) val += resid[(size_t)row * N + col];
        C[(size_t)row * N + col] = val;
      }
    }
  }
}

// ---------------------------------------------------------------------------
// L2-normalize rows of length 128 (keys). One wave per row, 4 rows per block.
// ---------------------------------------------------------------------------
__global__ __launch_bounds__(128) void l2norm_rows_kernel(float* __restrict__ k) {
  const int lane = threadIdx.x & 31;
  const int wave = threadIdx.x >> 5;
  const size_t row = (size_t)blockIdx.x * 4 + wave;
  float4* p = (float4*)(k + row * 128) + lane;
  float4 d = *p;
  float ss = d.x * d.x + d.y * d.y + d.z * d.z + d.w * d.w;
#pragma unroll
  for (int off = 16; off; off >>= 1) ss += __shfl_xor(ss, off, 32);
  const float scale = 1.0f / fmaxf(sqrtf(ss), 1e-12f);
  d.x *= scale; d.y *= scale; d.z *= scale; d.w *= scale;
  *p = d;
}

// ---------------------------------------------------------------------------
// beta/g: sigmoid(xn @ W + bias) for 16 heads each. One block per (b,t) row.
// ---------------------------------------------------------------------------
__global__ __launch_bounds__(256) void betag_kernel(
    const float* __restrict__ xn,
    const float* __restrict__ Wb, const float* __restrict__ bb,
    const float* __restrict__ Wg, const float* __restrict__ bg,
    float* __restrict__ beta, float* __restrict__ g) {
  __shared__ float row[Dc];
  const int r = blockIdx.x;
  const float* xr = xn + (size_t)r * Dc;
  for (int i = threadIdx.x; i < Dc; i += 256) row[i] = xr[i];
  __syncthreads();
  const int oi = threadIdx.x >> 3;  // 0..31
  const int s8 = threadIdx.x & 7;
  const int h = oi & 15;
  const float* W = (oi < 16) ? Wb : Wg;
  float sum = 0.f;
  for (int d = s8; d < Dc; d += 8) sum += row[d] * W[(size_t)d * Hc + h];
  sum += __shfl_xor(sum, 1, 32);
  sum += __shfl_xor(sum, 2, 32);
  sum += __shfl_xor(sum, 4, 32);
  if (s8 == 0) {
    const float bias = (oi < 16) ? bb[h] : bg[h];
    const float val = 1.0f / (1.0f + __expf(-(sum + bias)));
    if (oi < 16) beta[(size_t)r * Hc + h] = val;
    else         g[(size_t)r * Hc + h] = val;
  }
}

// ---------------------------------------------------------------------------
// Gated delta-rule scan. One block per (b,h); thread v owns state column
// S[:,v] in 128 VGPRs. k_t broadcast through LDS. T=2048 sequential steps.
// ---------------------------------------------------------------------------
__global__ __launch_bounds__(128) void delta_scan_kernel(
    const float* __restrict__ kbuf, const float* __restrict__ vbuf,
    const float* __restrict__ beta, const float* __restrict__ g,
    const float* __restrict__ S0, float* __restrict__ outbuf,
    float* __restrict__ Sout) {
  const int v = threadIdx.x;
  const int bh = blockIdx.x;
  const int b = bh / Hc, h = bh % Hc;
  __shared__ float sk[Kc];

  float s[Kc];
  const float* s0p = S0 + (size_t)bh * Kc * Vc + v;
#pragma unroll
  for (int kk = 0; kk < Kc; ++kk) s[kk] = s0p[(size_t)kk * Vc];

  for (int t = 0; t < Tc; ++t) {
    const size_t base = ((size_t)b * Tc + t) * Hc + h;
    const float* kp = kbuf + base * Kc;
    __syncthreads();               // all lanes done reading previous sk
    sk[v] = kp[v];
    const float vt = vbuf[base * Vc + v];
    const float bt = beta[base];
    const float gt = g[base];
    __syncthreads();

    float pred = 0.f;
#pragma unroll
    for (int kk = 0; kk < Kc; ++kk) pred = fmaf(s[kk], sk[kk], pred);
    const float u = bt * (vt - pred);
    float outv = 0.f;
#pragma unroll
    for (int kk = 0; kk < Kc; ++kk) {
      s[kk] = fmaf(gt, s[kk], sk[kk] * u);
      outv = fmaf(s[kk], sk[kk], outv);
    }
    outbuf[base * Vc + v] = outv;
  }

  float* sp = Sout + (size_t)bh * Kc * Vc + v;
#pragma unroll
  for (int kk = 0; kk < Kc; ++kk) sp[(size_t)kk * Vc] = s[kk];
}

// ---------------------------------------------------------------------------
extern "C" void kernel_launch(void* const* d_in, const int* in_sizes, int n_in,
                              void* d_out, int out_size, void* d_ws, size_t ws_size,
                              hipStream_t stream) {
  const float* x  = (const float*)d_in[0];
  const float* S0 = (const float*)d_in[1];
  const float* nw = (const float*)d_in[2];
  const float* Wk = (const float*)d_in[3];
  const float* Wv = (const float*)d_in[4];
  const float* Wo = (const float*)d_in[5];
  const float* Wb = (const float*)d_in[6];
  const float* bb = (const float*)d_in[7];
  const float* Wg = (const float*)d_in[8];
  const float* bg = (const float*)d_in[9];

  float* out  = (float*)d_out;
  float* Sout = out + (size_t)Bc * Tc * Dc;

  const size_t rows = (size_t)Bc * Tc;      // 4096
  float* ws   = (float*)d_ws;
  float* xn   = ws;                          // rows*Dc
  float* kb   = xn + rows * Dc;              // rows*Hc*Kc
  float* vb   = kb + rows * Hc * Kc;         // rows*Hc*Vc
  float* beta = vb + rows * Hc * Vc;         // rows*Hc
  float* gg   = beta + rows * Hc;            // rows*Hc
  float* ao   = xn;  // scan output aliases xn (xn dead after betag_kernel)

  rmsnorm_kernel<<<(uint32_t)rows, 256, 0, stream>>>(x, nw, xn, Dc);

  dim3 blk(256);
  dim3 g1(Hc * Kc / GEMM_BN, (uint32_t)rows / GEMM_BM);
  gemm_wmma_f32<<<g1, blk, 0, stream>>>(xn, Wk, nullptr, kb, (int)rows, Hc * Kc, Dc);
  gemm_wmma_f32<<<g1, blk, 0, stream>>>(xn, Wv, nullptr, vb, (int)rows, Hc * Vc, Dc);

  l2norm_rows_kernel<<<(uint32_t)(rows * Hc / 4), 128, 0, stream>>>(kb);
  betag_kernel<<<(uint32_t)rows, 256, 0, stream>>>(xn, Wb, bb, Wg, bg, beta, gg);

  delta_scan_kernel<<<Bc * Hc, 128, 0, stream>>>(kb, vb, beta, gg, S0, ao, Sout);

  dim3 g2(Dc / GEMM_BN, (uint32_t)rows / GEMM_BM);
  gemm_wmma_f32<<<g2, blk, 0, stream>>>(ao, Wo, x, out, (int)rows, Dc, Hc * Vc);
}